// RobustGIN_14199161880831
// MI455X (gfx1250) — compile-verified
//
#include <hip/hip_runtime.h>

#define NN 50000
#define EE 800000
#define DD 128
#define HH 128
#define GG 500
#define OO 10
#define BN_EPS 1e-5f

typedef __attribute__((ext_vector_type(2))) float v2f;
typedef __attribute__((ext_vector_type(8))) float v8f;

// ---------------------------------------------------------------------------
// 1) Gather embedding rows: x[i,:] = emb[x_ids[i],:]   (written into d_out tail)
// ---------------------------------------------------------------------------
__global__ __launch_bounds__(256) void gather_embed(
    const int* __restrict__ x_ids, const float* __restrict__ emb,
    float* __restrict__ x)
{
    int id = blockIdx.x * 256 + threadIdx.x;          // one float4 per thread
    if (id >= NN * (DD / 4)) return;
    int node = id >> 5;                               // DD/4 = 32 quads per node
    int q    = id & 31;
    const float4* e4 = reinterpret_cast<const float4*>(emb);
    float4* x4       = reinterpret_cast<float4*>(x);
    x4[(size_t)node * 32 + q] = e4[(size_t)x_ids[node] * 32 + q];
}

// ---------------------------------------------------------------------------
// 2) Edge scatter-add: agg[dst,:] += x[src,:]  (bandwidth-bound, f32 atomics)
// ---------------------------------------------------------------------------
__global__ __launch_bounds__(256) void edge_agg(
    const float* __restrict__ x, const int* __restrict__ src,
    const int* __restrict__ dst, float* __restrict__ agg)
{
    int id = blockIdx.x * 256 + threadIdx.x;          // one float4 per thread
    if (id >= EE * (HH / 4)) return;
    int e = id >> 5;
    int q = id & 31;
    int s = src[e];
    int d = dst[e];
    const float4* x4 = reinterpret_cast<const float4*>(x);
    float4 v = x4[(size_t)s * 32 + q];
    float* base = agg + (size_t)d * HH + q * 4;
    atomicAdd(base + 0, v.x);
    atomicAdd(base + 1, v.y);
    atomicAdd(base + 2, v.z);
    atomicAdd(base + 3, v.w);
}

// ---------------------------------------------------------------------------
// 3) GIN MLP on a 16-node panel using V_WMMA_F32_16X16X4_F32:
//    t = x + agg;  y = relu(BN(t@wa + ba));  hout = relu(y@wb + bb)
//    8 waves/block, wave w owns output columns [16w,16w+16).
// ---------------------------------------------------------------------------
#define LP (HH + 4)   // LDS row pitch (132) to spread banks

__global__ __launch_bounds__(256) void gin_mlp(
    const float* __restrict__ xin, const float* __restrict__ agg,
    const float* __restrict__ wa,  const float* __restrict__ ba,
    const float* __restrict__ g,   const float* __restrict__ be,
    const float* __restrict__ m,   const float* __restrict__ v,
    const float* __restrict__ wb,  const float* __restrict__ bb,
    float* __restrict__ hout)
{
    __shared__ float sA[16 * LP];
    __shared__ float sY[16 * LP];

    const int row0 = blockIdx.x * 16;
    const int tid  = threadIdx.x;

    // cooperative load of A = x + agg  (16 x 128)
    #pragma unroll
    for (int i = 0; i < 8; ++i) {
        int idx = tid + i * 256;            // 0..2047
        int r = idx >> 7, c = idx & 127;
        size_t gofs = (size_t)(row0 + r) * HH + c;
        sA[r * LP + c] = xin[gofs] + agg[gofs];
    }
    __syncthreads();

    const int lane = tid & 31;
    const int wave = tid >> 5;              // 0..7
    const int half = lane >> 4;             // K sub-block select
    const int l16  = lane & 15;
    const int n0   = wave * 16;
    const int col  = n0 + l16;

    // ---- GEMM1: sA(16x128) @ wa(128x128), tile cols [n0,n0+16) ----
    v8f acc = {};
    #pragma unroll 4
    for (int k = 0; k < HH; k += 4) {
        int ka = k + 2 * half;              // lanes 0-15: K=k,k+1 ; 16-31: K=k+2,k+3
        v2f a, b;
        a.x = sA[l16 * LP + ka + 0];
        a.y = sA[l16 * LP + ka + 1];
        b.x = wa[(size_t)(ka + 0) * HH + col];
        b.y = wa[(size_t)(ka + 1) * HH + col];
        acc = __builtin_amdgcn_wmma_f32_16x16x4_f32(
            false, a, false, b, (short)0, acc, false, false);
    }

    // epilogue: fused bias + BatchNorm(eval) + ReLU, store to sY
    {
        float scale = g[col] * rsqrtf(v[col] + BN_EPS);
        float shift = be[col] + (ba[col] - m[col]) * scale;
        #pragma unroll
        for (int r = 0; r < 8; ++r) {
            float val = acc[r] * scale + shift;    // (dot+ba-m)*scale+be
            sY[(r + 8 * half) * LP + col] = fmaxf(val, 0.0f);
        }
    }
    __syncthreads();

    // ---- GEMM2: sY(16x128) @ wb(128x128) ----
    v8f acc2 = {};
    #pragma unroll 4
    for (int k = 0; k < HH; k += 4) {
        int ka = k + 2 * half;
        v2f a, b;
        a.x = sY[l16 * LP + ka + 0];
        a.y = sY[l16 * LP + ka + 1];
        b.x = wb[(size_t)(ka + 0) * HH + col];
        b.y = wb[(size_t)(ka + 1) * HH + col];
        acc2 = __builtin_amdgcn_wmma_f32_16x16x4_f32(
            false, a, false, b, (short)0, acc2, false, false);
    }
    {
        float bias = bb[col];
        #pragma unroll
        for (int r = 0; r < 8; ++r) {
            int node = row0 + r + 8 * half;
            hout[(size_t)node * HH + col] = fmaxf(acc2[r] + bias, 0.0f);
        }
    }
}

// ---------------------------------------------------------------------------
// 4) global_add_pool of concat(h1,h2): hg[batch[i], :256] += [h1|h2][i,:]
// ---------------------------------------------------------------------------
__global__ __launch_bounds__(256) void pool_add(
    const float* __restrict__ h1, const float* __restrict__ h2,
    const int* __restrict__ batch, float* __restrict__ hg)
{
    int id = blockIdx.x * 256 + threadIdx.x;
    if (id >= NN * (2 * HH)) return;
    int node = id >> 8;
    int f    = id & 255;
    float val = (f < HH) ? h1[(size_t)node * HH + f]
                         : h2[(size_t)node * HH + (f - HH)];
    atomicAdd(&hg[(size_t)batch[node] * (2 * HH) + f], val);
}

// ---------------------------------------------------------------------------
// 5) Readout: hg2 = relu(hg @ wl1 + bl1)   [G,256]x[256,128]  (tiny)
// ---------------------------------------------------------------------------
__global__ __launch_bounds__(256) void readout1(
    const float* __restrict__ hg, const float* __restrict__ wl1,
    const float* __restrict__ bl1, float* __restrict__ hg2)
{
    int id = blockIdx.x * 256 + threadIdx.x;
    if (id >= GG * HH) return;
    int gi = id / HH, c = id % HH;
    float s = bl1[c];
    const float* row = hg + (size_t)gi * (2 * HH);
    #pragma unroll 8
    for (int k = 0; k < 2 * HH; ++k)
        s = fmaf(row[k], wl1[(size_t)k * HH + c], s);
    hg2[(size_t)gi * HH + c] = fmaxf(s, 0.0f);
}

// ---------------------------------------------------------------------------
// 6) Readout: out = hg2 @ wl2 + bl2   [G,128]x[128,10]  (tiny)
// ---------------------------------------------------------------------------
__global__ __launch_bounds__(256) void readout2(
    const float* __restrict__ hg2, const float* __restrict__ wl2,
    const float* __restrict__ bl2, float* __restrict__ out)
{
    int id = blockIdx.x * 256 + threadIdx.x;
    if (id >= GG * OO) return;
    int gi = id / OO, o = id % OO;
    float s = bl2[o];
    const float* row = hg2 + (size_t)gi * HH;
    #pragma unroll 8
    for (int k = 0; k < HH; ++k)
        s = fmaf(row[k], wl2[(size_t)k * OO + o], s);
    out[(size_t)gi * OO + o] = s;
}

// ---------------------------------------------------------------------------
extern "C" void kernel_launch(void* const* d_in, const int* in_sizes, int n_in,
                              void* d_out, int out_size, void* d_ws, size_t ws_size,
                              hipStream_t stream) {
    const int*   x_ids = (const int*)d_in[0];
    const int*   edges = (const int*)d_in[1];   // [2,E]: src = edges, dst = edges+E
    const int*   batch = (const int*)d_in[2];
    const float* emb   = (const float*)d_in[3];
    const float* w11 = (const float*)d_in[4],  *b11 = (const float*)d_in[5];
    const float* g1  = (const float*)d_in[6],  *be1 = (const float*)d_in[7];
    const float* m1  = (const float*)d_in[8],  *v1  = (const float*)d_in[9];
    const float* w12 = (const float*)d_in[10], *b12 = (const float*)d_in[11];
    const float* w21 = (const float*)d_in[12], *b21 = (const float*)d_in[13];
    const float* g2  = (const float*)d_in[14], *be2 = (const float*)d_in[15];
    const float* m2  = (const float*)d_in[16], *v2  = (const float*)d_in[17];
    const float* w22 = (const float*)d_in[18], *b22 = (const float*)d_in[19];
    const float* wl1 = (const float*)d_in[20], *bl1 = (const float*)d_in[21];
    const float* wl2 = (const float*)d_in[22], *bl2 = (const float*)d_in[23];

    float* out = (float*)d_out;
    float* x   = out + (size_t)GG * OO;          // node-logits slot doubles as x buffer

    float* ws  = (float*)d_ws;
    float* agg = ws;
    float* h1  = agg + (size_t)NN * HH;
    float* h2  = h1  + (size_t)NN * HH;
    float* hg  = h2  + (size_t)NN * HH;
    float* hg2 = hg  + (size_t)GG * (2 * HH);

    const int* src = edges;
    const int* dst = edges + EE;

    // x = emb[x_ids]
    gather_embed<<<(NN * 32 + 255) / 256, 256, 0, stream>>>(x_ids, emb, x);

    // layer 1
    hipMemsetAsync(agg, 0, (size_t)NN * HH * sizeof(float), stream);
    edge_agg<<<(EE * 32 + 255) / 256, 256, 0, stream>>>(x, src, dst, agg);
    gin_mlp<<<NN / 16, 256, 0, stream>>>(x, agg, w11, b11, g1, be1, m1, v1,
                                         w12, b12, h1);

    // layer 2
    hipMemsetAsync(agg, 0, (size_t)NN * HH * sizeof(float), stream);
    edge_agg<<<(EE * 32 + 255) / 256, 256, 0, stream>>>(h1, src, dst, agg);
    gin_mlp<<<NN / 16, 256, 0, stream>>>(h1, agg, w21, b21, g2, be2, m2, v2,
                                         w22, b22, h2);

    // pool + readout
    hipMemsetAsync(hg, 0, (size_t)GG * 2 * HH * sizeof(float), stream);
    pool_add<<<(NN * 256 + 255) / 256, 256, 0, stream>>>(h1, h2, batch, hg);
    readout1<<<(GG * HH + 255) / 256, 256, 0, stream>>>(hg, wl1, bl1, hg2);
    readout2<<<(GG * OO + 255) / 256, 256, 0, stream>>>(hg2, wl2, bl2, out);
}